// Layer_smoothing_filter_17686675324981
// MI455X (gfx1250) — compile-verified
//
#include <hip/hip_runtime.h>
#include <math.h>

// ---------------------------------------------------------------------------
// Layer smoothing filter, MI455X (gfx1250).
// out = r + sigmoid((|r|-mean)/std) * (sum(r) - n*r) / (n-1), per row of 1000.
//
// Bandwidth-bound: 262 MB total HBM traffic -> ~11.3 us floor at 23.3 TB/s.
// Design: 16 rows/block staged once into LDS via the Tensor Data Mover
// (iterate-mode descriptor, padded LDS stride 1028 floats for conflict-free
// WMMA A-fragment reads), three V_WMMA_F32_16X16X4_F32 accumulation chains
// (B = ones) compute the per-row sums of r, |r|, r^2 in the matrix pipe,
// then a fused second pass from LDS writes the result coalesced.
// ---------------------------------------------------------------------------

typedef __attribute__((ext_vector_type(2))) float    v2f;
typedef __attribute__((ext_vector_type(8))) float    v8f;
typedef __attribute__((ext_vector_type(4))) unsigned v4u;
typedef __attribute__((ext_vector_type(8))) int      v8i;
typedef __attribute__((ext_vector_type(4))) int      v4i;

#define N_COLS   1000
#define ROWS_PB  16
#define LDS_S    1028          // LDS row stride in floats (bank-conflict-free)
#define CHUNKS   (N_COLS / 4)  // 250 K=4 chunks per row
#define COEF     (1.0f / 999.0f)

__device__ __forceinline__ void wmma3(const v2f a, const v2f ones,
                                      v8f& cs, v8f& ca, v8f& cq)
{
    v2f aa; aa.x = fabsf(a.x);  aa.y = fabsf(a.y);
    v2f aq; aq.x = a.x * a.x;   aq.y = a.y * a.y;
    cs = __builtin_amdgcn_wmma_f32_16x16x4_f32(
             false, a,  false, ones, (short)0, cs, false, false);
    ca = __builtin_amdgcn_wmma_f32_16x16x4_f32(
             false, aa, false, ones, (short)0, ca, false, false);
    cq = __builtin_amdgcn_wmma_f32_16x16x4_f32(
             false, aq, false, ones, (short)0, cq, false, false);
}

__global__ void __launch_bounds__(256)
smooth_filter_kernel(const float* __restrict__ in, float* __restrict__ out)
{
    extern __shared__ float smem[];
    float* tile = smem;                       // [16][1028]
    float* red  = smem + ROWS_PB * LDS_S;     // [8 waves][3 quantities][16 rows]
    float* st   = red + 8 * 3 * ROWS_PB;      // [3][16]: sum, mean, inv_std

    const int tid  = threadIdx.x;
    const int lane = tid & 31;
    const int wv   = tid >> 5;

    const long long row0 = (long long)blockIdx.x * ROWS_PB;
    const float* gin  = in  + row0 * N_COLS;
    float*       gout = out + row0 * N_COLS;

    // ---- Stage 16 rows x 1000 f32 into LDS with the Tensor Data Mover ----
    // Iterate-mode 1D tile: 16 iterations, global += 1000 elems,
    // LDS += 1028 elems (padded stride). Wave 0 issues; TDM ignores EXEC.
    if (tid < 32) {
        unsigned lds_base = (unsigned)(unsigned long long)(size_t)tile;
        unsigned long long ga = (unsigned long long)(size_t)gin;

        v4u g0;
        g0[0] = 1u;                                   // count=1, no gather
        g0[1] = lds_base;                             // lds_addr (bytes)
        g0[2] = (unsigned)(ga & 0xFFFFFFFFu);         // global_addr[31:0]
        g0[3] = (unsigned)((ga >> 32) & 0x1FFFFFFu)   // global_addr[56:32]
              | (2u << 30);                           // type = 2 ("image")

        v8i g1;
        g1[0] = (2 << 16) | (1 << 19);                // data_size=4B, iterate_enable
        g1[1] = (int)((unsigned)N_COLS << 16);        // tensor_dim0[15:0]
        g1[2] = (int)(1u << 16);                      // tensor_dim1 = 1
        g1[3] = (int)((unsigned)N_COLS << 16);        // tile_dim0 = 1000
        g1[4] = 1;                                    // tile_dim1 = 1
        g1[5] = N_COLS;                               // tensor_dim0_stride
        g1[6] = (int)((unsigned)N_COLS << 16);        // tensor_dim1_stride[15:0]
        g1[7] = 0;

        v4i g2;
        g2[0] = 0;                                    // tensor_dim2 (unused)
        g2[1] = LDS_S;                                // lds_addr_increment (elems)
        g2[2] = N_COLS;                               // global_addr_increment (elems)
        g2[3] = (int)((ROWS_PB - 1u) << 16);          // iterate_count = 15 -> 16x

        v4i g3 = {0, 0, 0, 0};
        v8i g4 = {0, 0, 0, 0, 0, 0, 0, 0};            // extra block (clang-23 form)

        __builtin_amdgcn_tensor_load_to_lds(g0, g1, g2, g3, g4, 0);
        __builtin_amdgcn_s_wait_tensorcnt(0);
    }
    __syncthreads();

    // ---- Per-row sums of r, |r|, r^2 via WMMA f32 16x16x4, B = ones ------
    // A fragment for chunk c: lane L(<16): row L cols {4c,4c+1};
    //                         lane L+16 : row L cols {4c+2,4c+3}.
    // Stride 1028 => lane L touches banks 4L..4L+3: conflict-free b64 loads.
    // Scalar (readfirstlane) trip count -> s_cbranch loop, EXEC stays all-1s
    // for every WMMA. Unroll x2 keeps two ds_load_b64 in flight per pass.
    v8f cs = {}, ca = {}, cq = {};
    {
        const v2f ones = {1.0f, 1.0f};
        const int c0   = __builtin_amdgcn_readfirstlane(wv);
        const int base = (lane & 15) * LDS_S + ((lane >> 4) << 1);

        int c = c0;
        for (; c + 8 < CHUNKS; c += 16) {
            v2f a0 = *reinterpret_cast<const v2f*>(tile + base + 4 * c);
            v2f a1 = *reinterpret_cast<const v2f*>(tile + base + 4 * (c + 8));
            wmma3(a0, ones, cs, ca, cq);
            wmma3(a1, ones, cs, ca, cq);
        }
        if (c < CHUNKS) {
            v2f a0 = *reinterpret_cast<const v2f*>(tile + base + 4 * c);
            wmma3(a0, ones, cs, ca, cq);
        }
    }

    // D layout: VGPR v = row v (lanes 0-15) / row v+8 (lanes 16-31),
    // identical across N. Lanes 0 and 16 publish this wave's 16 row-partials.
    if (lane == 0 || lane == 16) {
        const int rbase = (lane >> 4) * 8;
        #pragma unroll
        for (int v = 0; v < 8; ++v) {
            red[(wv * 3 + 0) * ROWS_PB + rbase + v] = cs[v];
            red[(wv * 3 + 1) * ROWS_PB + rbase + v] = ca[v];
            red[(wv * 3 + 2) * ROWS_PB + rbase + v] = cq[v];
        }
    }
    __syncthreads();

    // ---- Combine the 8 wave-partials, derive per-row stats ---------------
    if (tid < ROWS_PB) {
        float s = 0.0f, sa = 0.0f, sq = 0.0f;
        #pragma unroll
        for (int w8 = 0; w8 < 8; ++w8) {
            s  += red[(w8 * 3 + 0) * ROWS_PB + tid];
            sa += red[(w8 * 3 + 1) * ROWS_PB + tid];
            sq += red[(w8 * 3 + 2) * ROWS_PB + tid];
        }
        const float mean = sa * (1.0f / (float)N_COLS);
        const float var  = (sq - sa * sa * (1.0f / (float)N_COLS)) * COEF; // ddof=1
        st[0 * ROWS_PB + tid] = s;
        st[1 * ROWS_PB + tid] = mean;
        st[2 * ROWS_PB + tid] = 1.0f / sqrtf(var);
    }
    __syncthreads();

    // ---- Fused elementwise pass from LDS, coalesced global stores --------
    for (int e = tid; e < ROWS_PB * N_COLS; e += 256) {
        const int row = e / N_COLS;
        const int col = e - row * N_COLS;
        const float r = tile[row * LDS_S + col];
        const float z = (fabsf(r) - st[1 * ROWS_PB + row]) * st[2 * ROWS_PB + row];
        const float w = 1.0f / (1.0f + __expf(-z));   // sigmoid, SCALE = 1
        gout[e] = r + w * COEF * (st[0 * ROWS_PB + row] - (float)N_COLS * r);
    }
}

extern "C" void kernel_launch(void* const* d_in, const int* in_sizes, int n_in,
                              void* d_out, int out_size, void* d_ws, size_t ws_size,
                              hipStream_t stream)
{
    (void)n_in; (void)d_ws; (void)ws_size; (void)out_size;
    const float* in = (const float*)d_in[0];
    float* out = (float*)d_out;

    const int rows   = in_sizes[0] / N_COLS;   // 32768
    const int blocks = rows / ROWS_PB;         // 2048

    const size_t shmem =
        (size_t)(ROWS_PB * LDS_S + 8 * 3 * ROWS_PB + 3 * ROWS_PB) * sizeof(float);

    smooth_filter_kernel<<<blocks, 256, shmem, stream>>>(in, out);
}